// EdgeDisplacement_5841155523157
// MI455X (gfx1250) — compile-verified
//
#include <hip/hip_runtime.h>
#include <stdint.h>

// ---------------------------------------------------------------------------
// Sobel.thin_edge (NMS + blob denoise), fused single-pass stencil for MI455X.
//
// Roofline: ~0 useful FLOPs/elt; fused = 1 read + 1 write = 8 B/elt fp32
// -> ceiling 23.3 TB/s / 8 B ~= 2.9 Telem/s. The NMS intermediate stays in
// LDS (320 KB/WGP); nothing but the input read and final write touches HBM.
//
// CDNA5-specific data path: input tiles are staged with the async
// global->LDS mover (global_load_async_to_lds_b128/b32, ASYNCcnt,
// s_wait_asynccnt) -- no VGPR round trip.
//
// This revision: Phase C now writes one 16-byte global_store_b128 (float4)
// per thread instead of 4 scalar b32 stores (stores are half the HBM
// traffic; 4x fewer vmem issues, full 128B coalescing per wave row group).
// Thread remap (row = tid>>3, col0 = (tid&7)*4) also lets contiguous
// nms_tile reads merge into wide ds loads.
//
// The harness passes only a flat element count; H/W are compile-time
// parameters (defaults below) and N*C is derived.
// ---------------------------------------------------------------------------

#ifndef EDGE_H
#define EDGE_H 1024
#endif
#ifndef EDGE_W
#define EDGE_W 1024
#endif

#define TILE      32
#define IN_DIM    (TILE + 4)        // 36 rows staged: output tile + halo 2
#define LDS_COLS  40                // staged cols: [tx0-4, tx0+36), 10 x b128
#define NMS_DIM   (TILE + 2)        // 34: NMS tile + halo 1 for denoise pass
#define LDS_PITCH 40                // floats per LDS row (conflict-free, 16B-aligned)
#define N_CHUNKS  (IN_DIM * (LDS_COLS / 4))   // 360 16-byte chunks per tile

__global__ __launch_bounds__(256) void sobel_thin_edge_fused(
    const float* __restrict__ in, float* __restrict__ out,
    int H, int W, int vec_ok)
{
  __shared__ float in_tile [IN_DIM  * LDS_PITCH];
  __shared__ float nms_tile[NMS_DIM * LDS_PITCH];

  const int tx0 = blockIdx.x * TILE;
  const int ty0 = blockIdx.y * TILE;

  const float* __restrict__ src = in  + (size_t)blockIdx.z * H * W;
  float* __restrict__       dst = out + (size_t)blockIdx.z * H * W;

  const int tid = threadIdx.y * 32 + threadIdx.x;   // 0..255, 8 wave32 waves

  // ---- Phase A: stage 36x40 zero-padded input window into LDS -------------
  // in_tile[r][c] = input[ty0-2+r][tx0-4+c]  (zeros outside the image).
  // Interior 16B chunks use the async data mover (ASYNCcnt). Boundary chunks
  // use per-element async b32 / ds-store zero; the address sets are disjoint
  // so ASYNCcnt/DScnt reordering cannot race.
  for (int i = tid; i < N_CHUNKS; i += 256) {
    const int r   = i / (LDS_COLS / 4);
    const int j   = i % (LDS_COLS / 4);
    const int gy  = ty0 - 2 + r;
    const int gx0 = tx0 - 4 + 4 * j;
    float* ldsp = &in_tile[r * LDS_PITCH + 4 * j];
    const bool rowin = (gy >= 0) & (gy < H);
    if (rowin & (vec_ok != 0) & (gx0 >= 0) & (gx0 + 3 < W)) {
      const unsigned lds_addr = (unsigned)(uintptr_t)ldsp;     // 16B-aligned
      const uint64_t gaddr    = (uint64_t)(uintptr_t)(src + (size_t)gy * W + gx0);
      asm volatile("global_load_async_to_lds_b128 %0, %1, off"
                   :: "v"(lds_addr), "v"(gaddr) : "memory");
    } else {
      #pragma unroll
      for (int k = 0; k < 4; ++k) {
        const int gx = gx0 + k;
        if (rowin && gx >= 0 && gx < W) {
          const unsigned lds_addr = (unsigned)(uintptr_t)(ldsp + k);
          const uint64_t gaddr    = (uint64_t)(uintptr_t)(src + (size_t)gy * W + gx);
          asm volatile("global_load_async_to_lds_b32 %0, %1, off"
                       :: "v"(lds_addr), "v"(gaddr) : "memory");
        } else {
          ldsp[k] = 0.0f;                                      // ds_store_b32
        }
      }
    }
  }
  // Drain this wave's async transfers, then barrier so every wave's LDS
  // writes are visible workgroup-wide.
  asm volatile("s_wait_asynccnt 0x0" ::: "memory");
  __syncthreads();

  // ---- Phase B: NMS into 34x34 LDS tile -----------------------------------
  // keep iff exists opposite pair (k, 8-k), k in {0..3}, both strictly < x.
  // nms(r,c) corresponds to image (ty0-1+r, tx0-1+c) -> in_tile[r+1][c+3].
  for (int i = tid; i < NMS_DIM * NMS_DIM; i += 256) {
    const int r = i / NMS_DIM;
    const int c = i % NMS_DIM;
    const float* p = &in_tile[(r + 1) * LDS_PITCH + (c + 3)];
    const float x  = p[0];
    const float n0 = p[-LDS_PITCH - 1], n1 = p[-LDS_PITCH], n2 = p[-LDS_PITCH + 1];
    const float n3 = p[-1],                                 n5 = p[1];
    const float n6 = p[ LDS_PITCH - 1], n7 = p[ LDS_PITCH], n8 = p[ LDS_PITCH + 1];
    const bool keep = ((n0 < x) & (n8 < x)) | ((n1 < x) & (n7 < x)) |
                      ((n2 < x) & (n6 < x)) | ((n3 < x) & (n5 < x));
    nms_tile[r * LDS_PITCH + c] = keep ? x : 0.0f;
  }
  __syncthreads();

  // ---- Phase C: blob denoise, one float4 store per thread -----------------
  // out = x_nms iff 20 * (sum of the 8 neighbors of x_nms) > 1.
  {
    const int r  = tid >> 3;          // 0..31
    const int c0 = (tid & 7) * 4;     // 0,4,...,28
    const float* p = &nms_tile[(r + 1) * LDS_PITCH + (c0 + 1)];
    float v[4];
    #pragma unroll
    for (int k = 0; k < 4; ++k) {
      const float x = p[k];
      const float s = p[k - LDS_PITCH - 1] + p[k - LDS_PITCH] + p[k - LDS_PITCH + 1]
                    + p[k - 1]             +                    p[k + 1]
                    + p[k + LDS_PITCH - 1] + p[k + LDS_PITCH] + p[k + LDS_PITCH + 1];
      v[k] = (20.0f * s > 1.0f) ? x : 0.0f;
    }
    const int gy = ty0 + r, gx0 = tx0 + c0;
    if (vec_ok && gy < H && gx0 + 3 < W) {
      *(float4*)(dst + (size_t)gy * W + gx0) = float4{v[0], v[1], v[2], v[3]};
    } else {
      #pragma unroll
      for (int k = 0; k < 4; ++k) {
        const int gx = gx0 + k;
        if (gy < H && gx < W)
          dst[(size_t)gy * W + gx] = v[k];
      }
    }
  }
}

extern "C" void kernel_launch(void* const* d_in, const int* in_sizes, int n_in,
                              void* d_out, int out_size, void* d_ws, size_t ws_size,
                              hipStream_t stream) {
  (void)n_in; (void)out_size; (void)d_ws; (void)ws_size;
  const float* x   = (const float*)d_in[0];
  float*       out = (float*)d_out;

  const long total = (long)in_sizes[0];
  int H = EDGE_H, W = EDGE_W;
  const long hw = (long)H * (long)W;
  long nc = total / hw;
  if (nc <= 0 || nc * hw != total) {      // unknown shape fallback: one wide row
    H = 1; W = (int)total; nc = 1;
  }
  const int vec_ok = (W % 4 == 0) ? 1 : 0;   // 16B alignment precondition

  const int tilesX = (W + TILE - 1) / TILE;
  const int tilesY = (H + TILE - 1) / TILE;

  dim3 block(32, 8);                              // 8 wave32 waves per block
  dim3 grid((unsigned)tilesX, (unsigned)tilesY, (unsigned)nc);
  sobel_thin_edge_fused<<<grid, block, 0, stream>>>(x, out, H, W, vec_ok);
}